// SpanEmbedding_21723944583200
// MI455X (gfx1250) — compile-verified
//
#include <hip/hip_runtime.h>
#include <hip/hip_bf16.h>

// Problem constants from the reference
#define BB 8
#define TT 2048
#define NN 512
#define HH 1024
#define WW 30

typedef float v2f __attribute__((ext_vector_type(2)));
typedef float v8f __attribute__((ext_vector_type(8)));

// ---------------------------------------------------------------------------
// Kernel 1: global attention logits  logits[b,t] = hiddens[b,t,:] . att_w + b
// One wave32 per token; float4 loads (512B/wave per load), shuffle reduction.
// ---------------------------------------------------------------------------
__global__ __launch_bounds__(256) void span_logits_kernel(
    const float* __restrict__ hiddens, const float* __restrict__ att_w,
    const float* __restrict__ att_b, float* __restrict__ logits) {
  int lane = threadIdx.x & 31;
  int wave = threadIdx.x >> 5;
  int token = blockIdx.x * 8 + wave;
  if (token >= BB * TT) return;

  const float4* hr = (const float4*)(hiddens + (size_t)token * HH);
  const float4* wr = (const float4*)att_w;
  float sum = 0.f;
#pragma unroll
  for (int i = 0; i < (HH / 4) / 32; ++i) {  // 8 iterations
    float4 h = hr[lane + 32 * i];
    float4 w = wr[lane + 32 * i];
    sum += h.x * w.x + h.y * w.y + h.z * w.z + h.w * w.w;
  }
#pragma unroll
  for (int off = 16; off > 0; off >>= 1) sum += __shfl_down(sum, off, 32);
  if (lane == 0) logits[token] = sum + att_b[0];
}

// ---------------------------------------------------------------------------
// Kernel 2: one block (4 wave32) per span.
//   coeff[2 x 32]:  row0 = mask/count (ave), row1 = masked-softmax weights
//   out[2 x H]   =  coeff x V[32 x H]   via V_WMMA_F32_16X16X4_F32
// A (16x4 f32, 2 VGPR): lane m<16 -> {K0,K1}, lane m>=16 -> {K2,K3}
// B (4x16 f32, 2 VGPR): lanes 0-15 -> rows K0,K1 at col; lanes 16-31 -> K2,K3
// D (16x16 f32, 8 VGPR): VGPR0 lanes0-15 = row M=0 (ave), VGPR1 = M=1 (self)
// ---------------------------------------------------------------------------
__global__ __launch_bounds__(128) void span_embed_kernel(
    const float* __restrict__ hiddens, const int* __restrict__ span_indices,
    const float* __restrict__ logits, float* __restrict__ out) {
  __shared__ float s_aw[32];  // ave coefficients (padded to 32)
  __shared__ float s_pw[32];  // softmax coefficients (padded to 32)

  const int blk = blockIdx.x;  // span id in [0, B*N)
  const int b = blk / NN;
  const int start = span_indices[blk * 2 + 0];
  const int end = span_indices[blk * 2 + 1];
  const int width = end - start;  // in [0, W-1]

  // --- first wave: exact AllenNLP masked softmax + ave weights into LDS ---
  if (threadIdx.x < 32) {
    int w = threadIdx.x;
    bool in30 = (w < WW);
    bool valid = in30 && (w <= width);
    float lg = valid ? logits[b * TT + start + w] : 0.f;
    float xv = valid ? lg : 0.f;  // v * mask (masked slots contribute exp(0))
    float m = in30 ? xv : -__builtin_inff();
#pragma unroll
    for (int off = 16; off > 0; off >>= 1) m = fmaxf(m, __shfl_xor(m, off, 32));
    float e = in30 ? expf(xv - m) : 0.f;
    float s = e;
#pragma unroll
    for (int off = 16; off > 0; off >>= 1) s += __shfl_xor(s, off, 32);
    float p = valid ? (e / s) : 0.f;  // softmax(v*mask) * mask
    float ps = p;
#pragma unroll
    for (int off = 16; off > 0; off >>= 1) ps += __shfl_xor(ps, off, 32);
    s_pw[w] = p / (ps + 1e-13f);                       // renormalize
    s_aw[w] = valid ? (1.f / (float)(width + 1)) : 0.f;  // masked mean
  }
  __syncthreads();

  const int lane = threadIdx.x & 31;
  const int wave = threadIdx.x >> 5;
  const int row = lane & 15;       // M row for A, column-in-tile for B/D
  const int hi = (lane >> 4) & 1;  // upper half-wave handles K+2,K+3

  // A operand: per-K-step coefficient pairs, reused across all 64 N-tiles
  v2f A[8];
  {
    const float* cp = (row == 1) ? s_pw : s_aw;
    float vf = (row < 2) ? 1.f : 0.f;  // rows 2..15 of A are zero
#pragma unroll
    for (int s8 = 0; s8 < 8; ++s8) {
      int w0 = 4 * s8 + 2 * hi;
      A[s8].x = cp[w0] * vf;
      A[s8].y = cp[w0 + 1] * vf;
    }
  }

  const float* vbase = hiddens + ((size_t)b * TT + start) * HH;
  float* obase = out + (size_t)blk * (2 * HH);

  // 16 N-tiles of 16 columns per wave (4 waves cover H=1024)
  for (int t = 0; t < 16; ++t) {
    int tb = (wave * 16 + t) * 16;
    int col = tb + row;
    v8f c = {0.f, 0.f, 0.f, 0.f, 0.f, 0.f, 0.f, 0.f};
#pragma unroll
    for (int s8 = 0; s8 < 8; ++s8) {
      int w0 = 4 * s8 + 2 * hi;
      int r0 = (w0 < WW) ? w0 : (WW - 1);  // pad rows 30,31 (coeff==0)
      int r1 = (w0 + 1 < WW) ? (w0 + 1) : (WW - 1);
      v2f bv;
      bv.x = vbase[(size_t)r0 * HH + col];
      bv.y = vbase[(size_t)r1 * HH + col];
      c = __builtin_amdgcn_wmma_f32_16x16x4_f32(false, A[s8], false, bv,
                                                (short)0, c, false, false);
    }
    if (!hi) {  // lanes 0-15: D row 0 = ave, D row 1 = selfatt
      obase[tb + row] = c[0];
      obase[HH + tb + row] = c[1];
    }
  }
}

extern "C" void kernel_launch(void* const* d_in, const int* in_sizes, int n_in,
                              void* d_out, int out_size, void* d_ws, size_t ws_size,
                              hipStream_t stream) {
  const float* hiddens = (const float*)d_in[0];
  const int* span_indices = (const int*)d_in[1];
  const float* att_w = (const float*)d_in[2];
  const float* att_b = (const float*)d_in[3];
  float* out = (float*)d_out;
  float* logits = (float*)d_ws;  // B*T floats = 64 KiB scratch

  span_logits_kernel<<<(BB * TT) / 8, 256, 0, stream>>>(hiddens, att_w, att_b,
                                                        logits);
  span_embed_kernel<<<BB * NN, 128, 0, stream>>>(hiddens, span_indices, logits,
                                                 out);
}